// VocabParallelEmbeddingWithLoRA_22376779612302
// MI455X (gfx1250) — compile-verified
//
#include <hip/hip_runtime.h>

typedef float v2f __attribute__((ext_vector_type(2)));
typedef float v8f __attribute__((ext_vector_type(8)));

#define RANK 16
#define ALDS_STRIDE 17   // pad to kill 16-word-stride LDS bank conflicts

__global__ __launch_bounds__(256) void lora_embed_wmma(
    const int* __restrict__ input_ids,
    const float* __restrict__ weight,
    const float* __restrict__ A_buffer,
    const float* __restrict__ B_buffer,
    const int* __restrict__ seg_indptr,
    const int* __restrict__ weight_indices,
    const float* __restrict__ scalings,
    float* __restrict__ out,
    int T, int dim, int vocab, int nseg_p1, int n_lora)
{
    __shared__ float a_lds[16 * ALDS_STRIDE];  // scaled LoRA-A rows [token][rank]
    __shared__ int   slot_lds[16];
    __shared__ int   id_lds[16];
    __shared__ int   mask_lds;

    const int tid = threadIdx.x;
    const int t0  = blockIdx.x * 16;

    // --- per-token segment lookup (searchsorted 'right' over tiny indptr) ---
    if (tid < 16) {
        int t = t0 + tid;
        if (t > T - 1) t = T - 1;              // tail clamp (reads only)
        int cnt = 0;
        for (int i = 0; i < nseg_p1; ++i) cnt += (seg_indptr[i] <= t) ? 1 : 0;
        int seg = cnt - 1;
        if (seg < 0) seg = 0;
        if (seg > nseg_p1 - 2) seg = nseg_p1 - 2;
        slot_lds[tid] = weight_indices[seg];
        id_lds[tid]   = input_ids[t];
    }
    __syncthreads();
    if (tid == 0) {
        int m = 0;
        for (int i = 0; i < 16; ++i) {
            int w = slot_lds[i];
            if (w != 0) m |= (1 << w);
        }
        mask_lds = m;
    }
    // --- gather + scale LoRA-A rows: thread (tok,rank) pair, 256 = 16x16 ---
    {
        int tok = tid >> 4;
        int r   = tid & 15;
        int w   = slot_lds[tok];
        float s = (w != 0) ? scalings[w] : 0.0f;   // slot 0 == inactive
        int id  = id_lds[tok];
        float av = A_buffer[((size_t)w * vocab + (size_t)id) * RANK + r];
        a_lds[tok * ALDS_STRIDE + r] = av * s;
    }
    __syncthreads();

    const int lane   = tid & 31;
    const int Mrow   = lane & 15;         // A-row (token) / B-col (dim) index
    const int hi     = lane >> 4;         // lane half
    const int koff   = hi * 2;            // K sublane offset within 4-chunk
    const int mySlot = slot_lds[Mrow];
    const int mask   = __builtin_amdgcn_readfirstlane(mask_lds);
    const int wave   = __builtin_amdgcn_readfirstlane((int)(threadIdx.x >> 5));

    // Unmasked A fragments for this lane, chunks kk = 0,4,8,12 (K=16 total).
    const float* ar = &a_lds[Mrow * ALDS_STRIDE + koff];
    const float ax0 = ar[0],  ay0 = ar[1];
    const float ax1 = ar[4],  ay1 = ar[5];
    const float ax2 = ar[8],  ay2 = ar[9];
    const float ax3 = ar[12], ay3 = ar[13];

    const int ntiles = dim >> 4;          // 16-wide dim tiles
    for (int nt = wave; nt < ntiles; nt += 8) {
        const int d0 = nt * 16;

        // C := base embedding tile, loaded directly in WMMA C-layout:
        // vgpr i -> row (i + 8*hi), col Mrow
        v8f c;
        #pragma unroll
        for (int i = 0; i < 8; ++i) {
            int row = i + hi * 8;
            c[i] = weight[(size_t)id_lds[row] * dim + d0 + Mrow];
        }

        // Accumulate LoRA contribution per active slot (scalar loop; usually 1)
        for (int l = 1; l < n_lora; ++l) {
            if (!(mask & (1 << l))) continue;
            const float sel = (mySlot == l) ? 1.0f : 0.0f;  // row mask for mixed tiles
            // Bop[r][d] = B_buffer[l][d][r]; lane loads 2 consecutive r's.
            const float* Bl = B_buffer + ((size_t)l * dim + (size_t)(d0 + Mrow)) * RANK + koff;
            v2f a, b;

            a[0] = ax0 * sel; a[1] = ay0 * sel;
            b[0] = Bl[0];     b[1] = Bl[1];
            c = __builtin_amdgcn_wmma_f32_16x16x4_f32(false, a, false, b, (short)0, c, false, false);

            a[0] = ax1 * sel; a[1] = ay1 * sel;
            b[0] = Bl[4];     b[1] = Bl[5];
            c = __builtin_amdgcn_wmma_f32_16x16x4_f32(false, a, false, b, (short)0, c, false, false);

            a[0] = ax2 * sel; a[1] = ay2 * sel;
            b[0] = Bl[8];     b[1] = Bl[9];
            c = __builtin_amdgcn_wmma_f32_16x16x4_f32(false, a, false, b, (short)0, c, false, false);

            a[0] = ax3 * sel; a[1] = ay3 * sel;
            b[0] = Bl[12];    b[1] = Bl[13];
            c = __builtin_amdgcn_wmma_f32_16x16x4_f32(false, a, false, b, (short)0, c, false, false);
        }

        #pragma unroll
        for (int i = 0; i < 8; ++i) {
            int row = i + hi * 8;
            if (t0 + row < T)
                out[(size_t)(t0 + row) * dim + d0 + Mrow] = c[i];
        }
    }
}

extern "C" void kernel_launch(void* const* d_in, const int* in_sizes, int n_in,
                              void* d_out, int out_size, void* d_ws, size_t ws_size,
                              hipStream_t stream) {
    const int*   input_ids      = (const int*)d_in[0];
    const float* weight         = (const float*)d_in[1];
    const float* A_buffer       = (const float*)d_in[2];
    const float* B_buffer       = (const float*)d_in[3];
    const int*   seg_indptr     = (const int*)d_in[4];
    const int*   weight_indices = (const int*)d_in[5];
    const float* scalings       = (const float*)d_in[6];
    float* out = (float*)d_out;

    const int T       = in_sizes[0];
    const int nseg_p1 = in_sizes[4];
    const int n_lora  = in_sizes[6];
    const int dim     = (int)((long long)in_sizes[3] / ((long long)n_lora * RANK)); // B: [L, DIM, RANK]
    const int vocab   = (int)((long long)in_sizes[1] / (long long)dim);             // W: [VOCAB, DIM]

    const int tiles = (T + 15) / 16;
    lora_embed_wmma<<<tiles, 256, 0, stream>>>(
        input_ids, weight, A_buffer, B_buffer,
        seg_indptr, weight_indices, scalings, out,
        T, dim, vocab, nseg_p1, n_lora);
}